// TAPE_87187836108927
// MI455X (gfx1250) — compile-verified
//
#include <hip/hip_runtime.h>
#include <math.h>

typedef __attribute__((ext_vector_type(2))) float v2f;
typedef __attribute__((ext_vector_type(8))) float v8f;

#define TAPE_N 2048
#define TAPE_D 512

// ---------------------------------------------------------------------------
// Kernel 1: per-batch masked intervals, reduction, and WMMA-based prefix sum.
// One block per batch. Wave 0 performs the scan with V_WMMA_F32_16X16X4_F32:
//   rowca = M x U          (U upper-triangular-inclusive)  -> row-wise prefix
//   P     = Lstrict x T + rowca  (T: row r broadcast of row-total r)
// giving the full inclusive prefix of a 256-element row-major 16x16 tile.
// ---------------------------------------------------------------------------
__global__ __launch_bounds__(256) void tape_pos_kernel(const float* __restrict__ time,
                                                       const int* __restrict__ data_size,
                                                       float* __restrict__ pos) {
    __shared__ float sm[TAPE_N];
    __shared__ float sred[256];
    __shared__ float s_inv;

    const int b   = blockIdx.x;
    const int tid = threadIdx.x;
    const int ds  = data_size[b];
    const float* tb = time + (size_t)b * TAPE_N;

    // masked intervals + local partial sum
    float lsum = 0.f;
#pragma unroll
    for (int j = 0; j < TAPE_N / 256; ++j) {
        int k = tid + j * 256;
        float v = 0.f;
        if (k > 0 && k < ds) v = tb[k] - tb[k - 1];
        sm[k] = v;
        lsum += v;
    }
    sred[tid] = lsum;
    __syncthreads();
    for (int s = 128; s > 0; s >>= 1) {
        if (tid < s) sred[tid] += sred[tid + s];
        __syncthreads();
    }
    if (tid == 0) {
        float sum_int = sred[0];
        if (sum_int == 0.f) sum_int = 1.f;
        float num_int = (float)ds - 1.f;
        if (num_int == 0.f) num_int = 1.f;
        s_inv = num_int / sum_int;
    }
    __syncthreads();

    if (tid < 32) {                       // wave 0 only: EXEC all-ones for WMMA
        const int lane = tid;
        const int row  = lane & 15;       // A: M index; B: N index; C/D: N index
        const int hi   = lane >> 4;       // which K-half this lane carries
        const float inv = s_inv;

        // B = U (upper-tri incl): B[K][j] = (K <= j), j == row for this lane.
        // VGPR0 carries K = 4k+2*hi, VGPR1 carries K = 4k+2*hi+1.
        v2f B0 = { (0 + 2*hi  <= row) ? 1.f : 0.f, (1 + 2*hi  <= row) ? 1.f : 0.f };
        v2f B1 = { (4 + 2*hi  <= row) ? 1.f : 0.f, (5 + 2*hi  <= row) ? 1.f : 0.f };
        v2f B2 = { (8 + 2*hi  <= row) ? 1.f : 0.f, (9 + 2*hi  <= row) ? 1.f : 0.f };
        v2f B3 = { (12 + 2*hi <= row) ? 1.f : 0.f, (13 + 2*hi <= row) ? 1.f : 0.f };
        // A' = Lstrict: A'[i][r] = (r < i), i == row.
        v2f L0 = { (0 + 2*hi  < row) ? 1.f : 0.f, (1 + 2*hi  < row) ? 1.f : 0.f };
        v2f L1 = { (4 + 2*hi  < row) ? 1.f : 0.f, (5 + 2*hi  < row) ? 1.f : 0.f };
        v2f L2 = { (8 + 2*hi  < row) ? 1.f : 0.f, (9 + 2*hi  < row) ? 1.f : 0.f };
        v2f L3 = { (12 + 2*hi < row) ? 1.f : 0.f, (13 + 2*hi < row) ? 1.f : 0.f };

        float carry = 0.f;
        float* pb = pos + (size_t)b * TAPE_N;

        for (int t = 0; t < 8; ++t) {
            // A slices: row 'row', columns 4k+2*hi, 4k+2*hi+1 (contiguous pairs)
            const float* tr = sm + t * 256 + row * 16 + 2 * hi;
            v2f a0 = *(const v2f*)(tr + 0);
            v2f a1 = *(const v2f*)(tr + 4);
            v2f a2 = *(const v2f*)(tr + 8);
            v2f a3 = *(const v2f*)(tr + 12);

            v8f c = {0.f, 0.f, 0.f, 0.f, 0.f, 0.f, 0.f, 0.f};
            c = __builtin_amdgcn_wmma_f32_16x16x4_f32(false, a0, false, B0, (short)0, c, false, false);
            c = __builtin_amdgcn_wmma_f32_16x16x4_f32(false, a1, false, B1, (short)0, c, false, false);
            c = __builtin_amdgcn_wmma_f32_16x16x4_f32(false, a2, false, B2, (short)0, c, false, false);
            c = __builtin_amdgcn_wmma_f32_16x16x4_f32(false, a3, false, B3, (short)0, c, false, false);

            // Row totals: rowtotal[r] = rowca[r][15]
            //   r<8 : component r, lane 15 ; r>=8 : component r-8, lane 31
            float rt0  = __shfl(c[0], 15, 32), rt1  = __shfl(c[1], 15, 32);
            float rt2  = __shfl(c[2], 15, 32), rt3  = __shfl(c[3], 15, 32);
            float rt4  = __shfl(c[4], 15, 32), rt5  = __shfl(c[5], 15, 32);
            float rt6  = __shfl(c[6], 15, 32), rt7  = __shfl(c[7], 15, 32);
            float rt8  = __shfl(c[0], 31, 32), rt9  = __shfl(c[1], 31, 32);
            float rt10 = __shfl(c[2], 31, 32), rt11 = __shfl(c[3], 31, 32);
            float rt12 = __shfl(c[4], 31, 32), rt13 = __shfl(c[5], 31, 32);
            float rt14 = __shfl(c[6], 31, 32), rt15 = __shfl(c[7], 31, 32);

            // B' slices: row K broadcast of rowtotal[K] across all N
            v2f Bp0 = { hi ? rt2  : rt0,  hi ? rt3  : rt1  };
            v2f Bp1 = { hi ? rt6  : rt4,  hi ? rt7  : rt5  };
            v2f Bp2 = { hi ? rt10 : rt8,  hi ? rt11 : rt9  };
            v2f Bp3 = { hi ? rt14 : rt12, hi ? rt15 : rt13 };

            c = __builtin_amdgcn_wmma_f32_16x16x4_f32(false, L0, false, Bp0, (short)0, c, false, false);
            c = __builtin_amdgcn_wmma_f32_16x16x4_f32(false, L1, false, Bp1, (short)0, c, false, false);
            c = __builtin_amdgcn_wmma_f32_16x16x4_f32(false, L2, false, Bp2, (short)0, c, false, false);
            c = __builtin_amdgcn_wmma_f32_16x16x4_f32(false, L3, false, Bp3, (short)0, c, false, false);

            float tile_total = __shfl(c[7], 31, 32);   // P[15][15] = tile sum

            // write pos = cumsum * inv + (k+1), masked
#pragma unroll
            for (int v = 0; v < 8; ++v) {
                int e = t * 256 + (v + 8 * hi) * 16 + row;   // C/D layout
                float val = (c[v] + carry) * inv + (float)(e + 1);
                if (e >= ds) val = 0.f;
                pb[e] = val;
            }
            carry += tile_total;
        }
    }
}

// ---------------------------------------------------------------------------
// Kernel 2: streaming out = x + tape. One float4 (two sin/cos pairs) / thread.
// Bandwidth-bound: 268 MB @ 23.3 TB/s ~ 11.5 us floor.
// ---------------------------------------------------------------------------
__global__ __launch_bounds__(256) void tape_apply_kernel(const float* __restrict__ x,
                                                         const float* __restrict__ pos,
                                                         float* __restrict__ out) {
    const unsigned tid = blockIdx.x * 256u + threadIdx.x;   // < b*n*d/4
    const unsigned dq  = tid & (TAPE_D / 4 - 1);            // float4 index in row
    const unsigned bn  = tid >> 7;                          // (b,n) flat index

    const float p = pos[bn];
    const float4 xv = ((const float4*)x)[tid];

    const float coef = -0.03597789207803197f;               // -2*ln(10000)/512
    const float j0 = (float)(2 * dq);
    const float w0 = expf(j0 * coef);
    const float w1 = expf((j0 + 1.0f) * coef);

    float s0, c0, s1, c1;
    sincosf(p * w0, &s0, &c0);
    sincosf(p * w1, &s1, &c1);

    float4 o;
    o.x = xv.x + s0;
    o.y = xv.y + c0;
    o.z = xv.z + s1;
    o.w = xv.w + c1;
    ((float4*)out)[tid] = o;
}

extern "C" void kernel_launch(void* const* d_in, const int* in_sizes, int n_in,
                              void* d_out, int out_size, void* d_ws, size_t ws_size,
                              hipStream_t stream) {
    const float* x         = (const float*)d_in[0];   // [32,2048,512] f32
    const float* time      = (const float*)d_in[1];   // [32,2048] f32
    const int*   data_size = (const int*)d_in[2];     // [32] i32
    float* out = (float*)d_out;
    float* pos = (float*)d_ws;                        // 32*2048 floats = 256 KB

    const int B = in_sizes[2];                        // 32
    tape_pos_kernel<<<B, 256, 0, stream>>>(time, data_size, pos);

    const int total4  = B * TAPE_N * (TAPE_D / 4);    // 8,388,608 threads
    const int nblocks = total4 / 256;                 // 32,768
    tape_apply_kernel<<<nblocks, 256, 0, stream>>>(x, pos, out);
}